// BinaryOperation_2388001816824
// MI455X (gfx1250) — compile-verified
//
#include <hip/hip_runtime.h>

// out[r,k] = x[r,k]*Lambda[k] + sum_l B[k,l]*y[r,l]
// N = 4*16*64*64*4 = 1,048,576 rows, L = 32.
// Strategy: fp32 WMMA (V_WMMA_F32_16X16X4_F32), C pre-loaded with x*Lambda so
// D = Y_tile x B^T + C is the final answer. Streamed x/y/out use non-temporal
// hints (384 MiB > 192 MB L2, single use). B/Lambda live in registers.

typedef __attribute__((ext_vector_type(2))) float v2f;
typedef __attribute__((ext_vector_type(8))) float v8f;

#define L_DIM   32
#define NROWS   (4 * 16 * 64 * 64 * 4)   // 1,048,576
#define NTILES  (NROWS / 16)             // 65,536 row-tiles of 16 rows
#define TILES_PER_WAVE 4

__global__ __launch_bounds__(256) void ssm_axbu_wmma(
    const float* __restrict__ x,
    const float* __restrict__ y,
    const float* __restrict__ Lambda,
    const float* __restrict__ B,
    float* __restrict__ out)
{
    const int lane = threadIdx.x & 31;
    const int n    = lane & 15;   // N (col) for B/C/D fragments, M (row) for A
    const int kh   = lane >> 4;   // K-half: adds +2 to K, +8 to M in C/D

    const int wavesPerBlock = blockDim.x >> 5;
    const int wid = blockIdx.x * wavesPerBlock + (threadIdx.x >> 5);

    // ---- Preload B^T fragments (whole 32x32 B stays in registers per wave).
    // WMMA-B operand, step s, col-tile t: lane holds
    //   B^T[K = 4s + 2*kh + {0,1}, N = 16t + n] = B[16t + n][4s + 2*kh .. +1]
    v2f bfrag[2][8];
#pragma unroll
    for (int t = 0; t < 2; ++t)
#pragma unroll
        for (int s = 0; s < 8; ++s)
            bfrag[t][s] = *(const v2f*)(B + (16 * t + n) * L_DIM + 4 * s + 2 * kh);

    const float lam0 = Lambda[n];
    const float lam1 = Lambda[16 + n];

    for (int i = 0; i < TILES_PER_WAVE; ++i) {
        const int tile = wid * TILES_PER_WAVE + i;
        if (tile >= NTILES) break;                 // wave-uniform: EXEC stays full
        const long row0 = (long)tile * 16;

        const float* __restrict__ xrow = x   + row0 * L_DIM;
        const float* __restrict__ yrow = y   + row0 * L_DIM;
        float*       __restrict__ orow = out + row0 * L_DIM;

        // ---- C init = x * Lambda  (C/D layout: VGPR g, lane -> M=g+8*kh, N=n)
        v8f c0, c1;
#pragma unroll
        for (int g = 0; g < 8; ++g) {
            const int r = g + 8 * kh;
            c0[g] = __builtin_nontemporal_load(xrow + r * L_DIM + n)      * lam0;
            c1[g] = __builtin_nontemporal_load(xrow + r * L_DIM + 16 + n) * lam1;
        }

        // ---- 8 K-steps of 16x16x4 fp32 WMMA, two column tiles share A.
        // A layout: lane -> M = n, K = 4s + 2*kh + {0,1}  (one b64 per step)
#pragma unroll
        for (int s = 0; s < 8; ++s) {
            v2f a = __builtin_nontemporal_load(
                (const v2f*)(yrow + (long)n * L_DIM + 4 * s + 2 * kh));
            c0 = __builtin_amdgcn_wmma_f32_16x16x4_f32(
                false, a, false, bfrag[0][s], (short)0, c0, false, false);
            c1 = __builtin_amdgcn_wmma_f32_16x16x4_f32(
                false, a, false, bfrag[1][s], (short)0, c1, false, false);
        }

        // ---- Store D (final result): coalesced 64B runs per half-wave.
#pragma unroll
        for (int g = 0; g < 8; ++g) {
            const int r = g + 8 * kh;
            __builtin_nontemporal_store(c0[g], orow + r * L_DIM + n);
            __builtin_nontemporal_store(c1[g], orow + r * L_DIM + 16 + n);
        }
    }
}

extern "C" void kernel_launch(void* const* d_in, const int* in_sizes, int n_in,
                              void* d_out, int out_size, void* d_ws, size_t ws_size,
                              hipStream_t stream) {
    const float* x      = (const float*)d_in[0];
    const float* y      = (const float*)d_in[1];
    const float* Lambda = (const float*)d_in[2];
    const float* B      = (const float*)d_in[3];
    float* out          = (float*)d_out;

    // 65,536 tiles / 4 per wave = 16,384 waves; 8 waves per 256-thread block.
    const int block = 256;
    const int grid  = (NTILES / TILES_PER_WAVE) / (block / 32);  // 2048
    ssm_axbu_wmma<<<grid, block, 0, stream>>>(x, y, Lambda, B, out);
}